// GAT_87342454931920
// MI455X (gfx1250) — compile-verified
//
#include <hip/hip_runtime.h>
#include <hip/hip_bf16.h>

typedef __attribute__((ext_vector_type(2))) float v2f;
typedef __attribute__((ext_vector_type(8))) float v8f;

static constexpr int kHid   = 128;
static constexpr int kHeads = 8;
static constexpr int kDh    = 16;

// ---------------------------------------------------------------------------
// C[n,128] = A[n,128] @ B + optional bias, f32 WMMA 16x16x4.
// layerB==0 : B is [128,128] row-major (encoder enc_W)
// layerB==1 : B is [8,128,16] (W_stack[ell]); logical col j -> head j>>4, f j&15
// Block = 256 threads = 8 waves; wave w handles 16-col tile w (== head w),
// block handles one 16-row strip. WMMA needs EXEC all-1s: branches are uniform.
// ---------------------------------------------------------------------------
__global__ __launch_bounds__(256) void gemm128_wmma(
    const float* __restrict__ A, const float* __restrict__ B,
    const float* __restrict__ bias, float* __restrict__ C,
    int nrows, int layerB)
{
  const int wave = threadIdx.x >> 5;
  const int lane = threadIdx.x & 31;
  const int row0 = blockIdx.x << 4;
  if (row0 >= nrows) return;                 // uniform per block
  const int m    = lane & 15;                // A row within tile
  const int nc   = lane & 15;                // C col within tile
  const int kh   = (lane >> 4) << 1;         // 0 (lanes 0-15) or 2 (lanes 16-31)
  const int col0 = wave << 4;

  const float* Ar = A + (size_t)(row0 + m) * kHid;
  const float* Bp;
  int bstride;
  if (layerB) { Bp = B + (size_t)wave * kHid * kDh + nc; bstride = kDh;  }
  else        { Bp = B + col0 + nc;                      bstride = kHid; }

  v8f acc = {};
  for (int k0 = 0; k0 < kHid; k0 += 4) {
    v2f av, bv;
    av.x = Ar[k0 + kh];
    av.y = Ar[k0 + kh + 1];
    bv.x = Bp[(size_t)(k0 + kh) * bstride];
    bv.y = Bp[(size_t)(k0 + kh + 1) * bstride];
    // D = A(16x4) * B(4x16) + C   -> v_wmma_f32_16x16x4_f32
    acc = __builtin_amdgcn_wmma_f32_16x16x4_f32(
        /*neg_a=*/false, av, /*neg_b=*/false, bv,
        /*c_mod=*/(short)0, acc, /*reuse_a=*/false, /*reuse_b=*/false);
  }

  const float bb = bias ? bias[col0 + nc] : 0.0f;
#pragma unroll
  for (int v = 0; v < 8; ++v) {
    const int r = row0 + v + ((lane >> 4) << 3);   // M = v (+8 for hi half)
    C[(size_t)r * kHid + col0 + nc] = acc[v] + bb;
  }
}

// alpha_src/alpha_dst per (node, head): dot(h[node, h*16:+16], a[h, 0:16 / 16:32])
__global__ void alpha_kernel(const float* __restrict__ h_all,
                             const float* __restrict__ a,
                             float* __restrict__ asrc, float* __restrict__ adst,
                             int n)
{
  const int idx = blockIdx.x * blockDim.x + threadIdx.x;
  if (idx >= n * kHeads) return;
  const int node = idx >> 3, hh = idx & 7;
  const float* hv = h_all + (size_t)node * kHid + hh * kDh;
  const float* av = a + hh * 2 * kDh;
  float s = 0.0f, d = 0.0f;
#pragma unroll
  for (int f = 0; f < kDh; ++f) {
    const float v = hv[f];
    s += v * av[f];
    d += v * av[kDh + f];
  }
  asrc[idx] = s;
  adst[idx] = d;
}

__global__ void zero_kernel(float* __restrict__ p, size_t len)
{
  size_t i = (size_t)blockIdx.x * blockDim.x + threadIdx.x;
  const size_t stride = (size_t)gridDim.x * blockDim.x;
  for (; i < len; i += stride) p[i] = 0.0f;
}

// One thread per (edge, head): softmax weight + scatter-add into rowsum/agg.
// agg (25.6 MB) and rowsum (1.6 MB) are L2-resident; unsafeAtomicAdd lowers to
// hardware global_atomic_add_f32 (no CAS loop).
__global__ void edge_kernel(const int* __restrict__ row, const int* __restrict__ col,
                            const float* __restrict__ evals,
                            const float* __restrict__ asrc, const float* __restrict__ adst,
                            const float* __restrict__ h_all,
                            float* __restrict__ rowsum, float* __restrict__ agg,
                            int E)
{
  const int idx = blockIdx.x * blockDim.x + threadIdx.x;
  if (idx >= E * kHeads) return;
  const int e  = idx >> 3;
  const int hh = idx & 7;
  const int r = row[e], c = col[e];
  const float logit = evals[e] * (asrc[r * kHeads + hh] + adst[c * kHeads + hh]);
  const float lr = logit > 0.0f ? logit : 0.2f * logit;   // leaky_relu(0.2)
  const float w = expf(lr);
  unsafeAtomicAdd(&rowsum[r * kHeads + hh], w);
  const float* hv = h_all + (size_t)c * kHid + hh * kDh;
  float* av = agg + (size_t)r * kHid + hh * kDh;
#pragma unroll
  for (int f = 0; f < kDh; ++f) unsafeAtomicAdd(&av[f], w * hv[f]);
}

// out = agg / rowsum; mode 0: ELU (layer 0, not last); mode 1: + residual.
__global__ void finalize_kernel(const float* __restrict__ agg,
                                const float* __restrict__ rowsum,
                                const float* __restrict__ resid,
                                float* __restrict__ out, int n, int mode)
{
  const int idx = blockIdx.x * blockDim.x + threadIdx.x;
  if (idx >= n * kHid) return;
  const int node = idx >> 7;
  const int hh   = (idx & 127) >> 4;
  float v = agg[idx] / rowsum[node * kHeads + hh];
  if (mode == 0) v = v > 0.0f ? v : (expf(v) - 1.0f);   // elu, alpha=1
  else           v += resid[idx];
  out[idx] = v;
}

extern "C" void kernel_launch(void* const* d_in, const int* in_sizes, int n_in,
                              void* d_out, int out_size, void* d_ws, size_t ws_size,
                              hipStream_t stream)
{
  const float* x       = (const float*)d_in[0];
  const int*   eidx    = (const int*)  d_in[1];
  const float* evals   = (const float*)d_in[2];
  const float* enc_W   = (const float*)d_in[3];
  const float* enc_b   = (const float*)d_in[4];
  const float* W_stack = (const float*)d_in[5];
  const float* a_stack = (const float*)d_in[6];

  const int n = in_sizes[0] / kHid;   // 50000 (divisible by 16)
  const int E = in_sizes[2];          // 850000
  const int* row = eidx;              // edge_idx[0,:]
  const int* col = eidx + E;          // edge_idx[1,:]

  float* ws     = (float*)d_ws;
  float* bufA   = ws;                         // n*128 : enc output, then agg
  float* rowsum = ws + (size_t)n * 128;       // n*8   : adjacent to bufA -> one zero pass
  float* bufB   = ws + (size_t)n * 136;       // n*128 : layer-0 output (residual source)
  float* bufH   = ws + (size_t)n * 264;       // n*128 : per-layer h_all
  float* asrc   = ws + (size_t)n * 392;       // n*8
  float* adst   = ws + (size_t)n * 400;       // n*8

  const dim3 blk(256);
  const int gemmGrid = n / 16;                // 3125 strips
  const int naThreads = n * kHeads;
  const int neThreads = E * kHeads;           // 6.8M, fits int
  const int nfThreads = n * kHid;

  // Encoder: bufA = x @ enc_W + enc_b
  gemm128_wmma<<<gemmGrid, blk, 0, stream>>>(x, enc_W, enc_b, bufA, n, 0);

  const float* xin = bufA;
  for (int ell = 0; ell < 2; ++ell) {
    const float* Wl = W_stack + (size_t)ell * kHeads * kHid * kDh;
    const float* al = a_stack + (size_t)ell * kHeads * 2 * kDh;

    // h_all = xin @ Wcat  (column tile w == head w)
    gemm128_wmma<<<gemmGrid, blk, 0, stream>>>(xin, Wl, nullptr, bufH, n, 1);
    alpha_kernel<<<(naThreads + 255) / 256, blk, 0, stream>>>(bufH, al, asrc, adst, n);

    // bufA (xin for layer 0) is dead past this point -> recycle as agg
    zero_kernel<<<2048, blk, 0, stream>>>(bufA, (size_t)n * 136);  // agg + rowsum

    edge_kernel<<<(neThreads + 255) / 256, blk, 0, stream>>>(
        row, col, evals, asrc, adst, bufH, rowsum, bufA, E);

    if (ell == 0) {
      finalize_kernel<<<(nfThreads + 255) / 256, blk, 0, stream>>>(
          bufA, rowsum, nullptr, bufB, n, /*mode=*/0);
      xin = bufB;
    } else {
      finalize_kernel<<<(nfThreads + 255) / 256, blk, 0, stream>>>(
          bufA, rowsum, bufB, (float*)d_out, n, /*mode=*/1);
    }
  }
}